// GraphReinforceAgent_33887291965745
// MI455X (gfx1250) — compile-verified
//
#include <hip/hip_runtime.h>
#include <math.h>

typedef float v2f __attribute__((ext_vector_type(2)));
typedef float v8f __attribute__((ext_vector_type(8)));

#define HDIM 128
#define NACT 16

// ---------------------------------------------------------------------------
// init workspace: deg=1 (self loop), sfi=0 (interleaved g0/g1), colsum=0, sums=0
// ---------------------------------------------------------------------------
__global__ void k_init(float* __restrict__ deg, float* __restrict__ sfi,
                       float* __restrict__ colsum, float* __restrict__ sums, int n) {
    int i = blockIdx.x * blockDim.x + threadIdx.x;
    if (i < n) deg[i] = 1.0f;
    if (i < 2 * n) sfi[i] = 0.0f;
    if (i < HDIM) colsum[i] = 0.0f;
    if (i < 2) sums[i] = 0.0f;
}

// ---------------------------------------------------------------------------
// deg[dst] += 1 per edge
// ---------------------------------------------------------------------------
__global__ void k_deg(const int* __restrict__ dst, float* __restrict__ deg, int e) {
    int i = blockIdx.x * blockDim.x + threadIdx.x;
    if (i < e) atomicAdd(&deg[dst[i]], 1.0f);
}

// ---------------------------------------------------------------------------
// dis = 1/sqrt(deg)
// ---------------------------------------------------------------------------
__global__ void k_dis(const float* __restrict__ deg, float* __restrict__ dis, int n) {
    int i = blockIdx.x * blockDim.x + threadIdx.x;
    if (i < n) dis[i] = 1.0f / sqrtf(deg[i]);
}

// ---------------------------------------------------------------------------
// Edge scatter: only 2 floats per edge (linearity of the GCN layer).
// Interleaved so both atomics land on the same cacheline.
// ---------------------------------------------------------------------------
__global__ void k_edge(const int* __restrict__ src, const int* __restrict__ dst,
                       const float* __restrict__ feats, const float* __restrict__ dis,
                       float* __restrict__ sfi, int e) {
    int i = blockIdx.x * blockDim.x + threadIdx.x;
    if (i < e) {
        int s = src[i];
        int d = dst[i];
        float nrm = dis[s] * dis[d];
        float2 f = ((const float2*)feats)[s];
        atomicAdd(&sfi[2 * d],     nrm * f.x);
        atomicAdd(&sfi[2 * d + 1], nrm * f.y);
    }
}

// ---------------------------------------------------------------------------
// Node stage with WMMA F32 16x16x4:
//   x[n,c] = relu(g0[n]*w0[c] + g1[n]*w1[c] + 1*gcn_b[c] + 0)
// A (16x4): K0=g0, K1=g1, K2=1 (bias row), K3=0
// B (4x16): K0=w0[c], K1=w1[c], K2=gcn_b[c], K3=0  -- held in registers
// Grid-stride over 16-node tiles; per-lane register accumulation of column
// sums (channel assignment is tile-invariant), sum and sumsq.
// x is never materialized.
// ---------------------------------------------------------------------------
__global__ void __launch_bounds__(256)
k_node_wmma(const float2* __restrict__ sfi, const float* __restrict__ dis,
            const float* __restrict__ feats, const float* __restrict__ gcn_w,
            const float* __restrict__ gcn_b, float* __restrict__ colsum,
            float* __restrict__ sums, int n, int ntiles) {
    __shared__ float lcol[HDIM];
    __shared__ float lsumArr[8];
    __shared__ float lsqArr[8];

    int t = threadIdx.x;
    if (t < HDIM) lcol[t] = 0.0f;
    __syncthreads();

    int lane = t & 31;
    int wave = t >> 5;
    int cl   = lane & 15;

    // Preload all 8 B tiles into registers (loop-invariant).
    float b0[8], b1[8];
    #pragma unroll
    for (int ct = 0; ct < 8; ++ct) {
        int c = ct * 16 + cl;
        if (lane < 16) { b0[ct] = gcn_w[c]; b1[ct] = gcn_w[HDIM + c]; }  // K0,K1
        else           { b0[ct] = gcn_b[c]; b1[ct] = 0.0f;            }  // K2,K3
    }

    float colacc[8];
    #pragma unroll
    for (int ct = 0; ct < 8; ++ct) colacc[ct] = 0.0f;
    float lsum = 0.0f, lsq = 0.0f;

    int gwave  = blockIdx.x * 8 + wave;
    int nwaves = gridDim.x * 8;

    for (int tile = gwave; tile < ntiles; tile += nwaves) {
        int base = tile * 16;

        // ---- build A (16x4 f32: VGPR0 = K0|K2, VGPR1 = K1|K3 by lane halves)
        v2f A;
        if (lane < 16) {
            int node = base + lane;
            float a0 = 0.0f, a1 = 0.0f;
            if (node < n) {
                float dd = dis[node];
                float d2 = dd * dd;
                float2 f = ((const float2*)feats)[node];
                float2 s = sfi[node];
                a0 = s.x + d2 * f.x;
                a1 = s.y + d2 * f.y;
            }
            A[0] = a0; A[1] = a1;
        } else {
            A[0] = 1.0f;   // K=2 -> bias row
            A[1] = 0.0f;   // K=3
        }

        if (base + 16 <= n) {             // wave-uniform fast path (no guards)
            #pragma unroll
            for (int ct = 0; ct < 8; ++ct) {
                v2f B; B[0] = b0[ct]; B[1] = b1[ct];
                v8f C = {};
                v8f D = __builtin_amdgcn_wmma_f32_16x16x4_f32(
                            false, A, false, B, (short)0, C, false, false);
                float cp = 0.0f;
                #pragma unroll
                for (int i = 0; i < 8; ++i) {
                    float x = fmaxf(D[i], 0.0f);
                    cp  += x;
                    lsq  = fmaf(x, x, lsq);
                }
                colacc[ct] += cp;
                lsum       += cp;
            }
        } else {                           // partial tail tile (rare)
            #pragma unroll
            for (int ct = 0; ct < 8; ++ct) {
                v2f B; B[0] = b0[ct]; B[1] = b1[ct];
                v8f C = {};
                v8f D = __builtin_amdgcn_wmma_f32_16x16x4_f32(
                            false, A, false, B, (short)0, C, false, false);
                float cp = 0.0f;
                #pragma unroll
                for (int i = 0; i < 8; ++i) {
                    int node = base + i + ((lane >= 16) ? 8 : 0);
                    float x = (node < n) ? fmaxf(D[i], 0.0f) : 0.0f;
                    cp  += x;
                    lsq  = fmaf(x, x, lsq);
                }
                colacc[ct] += cp;
                lsum       += cp;
            }
        }
    }

    // ---- per-wave epilogue: combine node-halves sharing a channel, LDS acc
    #pragma unroll
    for (int ct = 0; ct < 8; ++ct) {
        float cp = colacc[ct];
        cp += __shfl_xor(cp, 16, 32);
        if (lane < 16) atomicAdd(&lcol[ct * 16 + cl], cp);
    }

    #pragma unroll
    for (int off = 16; off > 0; off >>= 1) {
        lsum += __shfl_xor(lsum, off, 32);
        lsq  += __shfl_xor(lsq,  off, 32);
    }
    if (lane == 0) { lsumArr[wave] = lsum; lsqArr[wave] = lsq; }
    __syncthreads();

    if (t == 0) {
        float s = 0.0f, q = 0.0f;
        #pragma unroll
        for (int w = 0; w < 8; ++w) { s += lsumArr[w]; q += lsqArr[w]; }
        atomicAdd(&sums[0], s);
        atomicAdd(&sums[1], q);
    }
    if (t < HDIM) atomicAdd(&colsum[t], lcol[t]);
}

// ---------------------------------------------------------------------------
// Head: closed-form LayerNorm+pool, fc1 (628x128), fc2 (128x16), log_softmax.
// Single block, 128 threads.
// ---------------------------------------------------------------------------
__global__ void k_head(const float* __restrict__ colsum, const float* __restrict__ sums,
                       const float* __restrict__ esn,
                       const float* __restrict__ ln_w, const float* __restrict__ ln_b,
                       const float* __restrict__ fc1_w, const float* __restrict__ fc1_b,
                       const float* __restrict__ fc2_w, const float* __restrict__ fc2_b,
                       float* __restrict__ out, int n, int r) {
    __shared__ float z[HDIM + 512];
    __shared__ float z2[HDIM];
    __shared__ float lg[NACT];

    int t = threadIdx.x;
    float cnt   = (float)n * (float)HDIM;
    float mean  = sums[0] / cnt;
    float var   = sums[1] / cnt - mean * mean;
    float scale = 1.0f / (sqrtf(var) + 1e-5f);
    float nf    = (float)n;

    if (t < HDIM)
        z[t] = ln_w[t] * (colsum[t] - nf * mean) * scale + nf * ln_b[t];
    for (int k = t; k < r; k += 128)
        z[HDIM + k] = esn[k];
    __syncthreads();

    int kmax = HDIM + r;
    float acc = fc1_b[t];
    for (int k = 0; k < kmax; ++k)
        acc += z[k] * fc1_w[k * HDIM + t];
    z2[t] = fmaxf(acc, 0.0f);
    __syncthreads();

    if (t < NACT) {
        float a = fc2_b[t];
        #pragma unroll 8
        for (int k = 0; k < HDIM; ++k)
            a += z2[k] * fc2_w[k * NACT + t];
        lg[t] = a;
    }
    __syncthreads();

    if (t < NACT) {
        float m = -INFINITY;
        #pragma unroll
        for (int k = 0; k < NACT; ++k) m = fmaxf(m, lg[k]);
        float s = 0.0f;
        #pragma unroll
        for (int k = 0; k < NACT; ++k) s += expf(lg[k] - m);
        out[t] = lg[t] - m - logf(s);
    }
}

// ---------------------------------------------------------------------------
extern "C" void kernel_launch(void* const* d_in, const int* in_sizes, int n_in,
                              void* d_out, int out_size, void* d_ws, size_t ws_size,
                              hipStream_t stream) {
    const float* node_feats = (const float*)d_in[0];
    const int*   edge_index = (const int*)  d_in[1];
    const float* esn        = (const float*)d_in[2];
    const float* gcn_w      = (const float*)d_in[3];
    const float* gcn_b      = (const float*)d_in[4];
    const float* ln_w       = (const float*)d_in[5];
    const float* ln_b       = (const float*)d_in[6];
    const float* fc1_w      = (const float*)d_in[7];
    const float* fc1_b      = (const float*)d_in[8];
    const float* fc2_w      = (const float*)d_in[9];
    const float* fc2_b      = (const float*)d_in[10];
    float* out = (float*)d_out;

    int n = in_sizes[0] / 2;   // nodes
    int e = in_sizes[1] / 2;   // edges
    int r = in_sizes[2];       // reservoir size

    float* ws     = (float*)d_ws;
    float* deg    = ws;
    float* dis    = ws + (size_t)n;
    float* sfi    = ws + 2 * (size_t)n;     // interleaved g0/g1, 2n floats
    float* colsum = ws + 4 * (size_t)n;
    float* sums   = colsum + HDIM;

    const int* src = edge_index;
    const int* dst = edge_index + e;

    const int tb = 256;
    k_init<<<(2 * n + tb - 1) / tb, tb, 0, stream>>>(deg, sfi, colsum, sums, n);
    k_deg <<<(e + tb - 1) / tb, tb, 0, stream>>>(dst, deg, e);
    k_dis <<<(n + tb - 1) / tb, tb, 0, stream>>>(deg, dis, n);
    k_edge<<<(e + tb - 1) / tb, tb, 0, stream>>>(src, dst, node_feats, dis, sfi, e);

    int ntiles = (n + 15) / 16;
    int nblocks = (ntiles + 7) / 8;
    if (nblocks > 256) nblocks = 256;      // grid-stride: ~3 tiles per wave
    k_node_wmma<<<nblocks, 256, 0, stream>>>((const float2*)sfi, dis, node_feats,
                                             gcn_w, gcn_b, colsum, sums, n, ntiles);

    k_head<<<1, 128, 0, stream>>>(colsum, sums, esn, ln_w, ln_b,
                                  fc1_w, fc1_b, fc2_w, fc2_b, out, n, r);
}